// RGCNConv3_1460288881434
// MI455X (gfx1250) — compile-verified
//
#include <hip/hip_runtime.h>

// Problem constants (match reference)
#define N_NODES 512
#define F_DIM   64
#define DG      16
#define FE      32
#define NM1     511
#define E_EDGES (N_NODES * NM1)        // 261632
#define ROW_STRIDE (2 * F_DIM + FE)    // 160 floats per output row

typedef float v2f __attribute__((ext_vector_type(2)));
typedef float v4f __attribute__((ext_vector_type(4)));
typedef float v8f __attribute__((ext_vector_type(8)));

// ---------------------------------------------------------------------------
// Kernel 1: gather sender/receiver features into out[:, 0:128].
// One wave (32 lanes) per edge. Lanes 0..15 write senders[0..63] as float4s,
// lanes 16..31 write receivers[0..63]. Per wave this is one contiguous 512B
// region of the output row -> fully coalesced B128 stores.
// ---------------------------------------------------------------------------
__global__ __launch_bounds__(256) void gather_kernel(
    const float* __restrict__ nodes,   // [N_NODES, F_DIM]
    float* __restrict__ out)           // [E_EDGES, ROW_STRIDE]
{
    const unsigned waves_per_blk = blockDim.x >> 5;
    const unsigned wave = blockIdx.x * waves_per_blk + (threadIdx.x >> 5);
    const unsigned lane = threadIdx.x & 31;
    const unsigned waves_total = gridDim.x * waves_per_blk;

    const unsigned half = lane >> 4;       // 0 = sender segment, 1 = receiver
    const unsigned q    = lane & 15;       // float4 index within segment

    for (unsigned e = wave; e < E_EDGES; e += waves_total) {
        const unsigned i = e / NM1;                // receiver node
        const unsigned k = e - i * NM1;
        const unsigned j = k + (k >= i ? 1u : 0u); // sender node
        const unsigned node = half ? i : j;

        const v4f val = *(const v4f*)(nodes + node * F_DIM + q * 4);
        __builtin_nontemporal_store(
            val, (v4f*)(out + (size_t)e * ROW_STRIDE + half * F_DIM + q * 4));
    }
}

// ---------------------------------------------------------------------------
// Kernel 2: edges_g = global_attr[E,16] @ edge_attr[16,32] into out[:,128:160]
// via V_WMMA_F32_16X16X4_F32. One wave per 16-edge tile; K=16 done as four
// accumulating k=4 steps; FE=32 done as two N-tiles of 16.
//
// A 16x4 f32 layout: lanes 0-15 hold M=lane, {K=k0,k0+1}; lanes 16-31 hold
// M=lane-16, {K=k0+2,k0+3}.  B 4x16 mirrors with N across lanes.
// C/D 16x16 f32 layout: VGPR v -> M=v (lanes 0-15) / M=v+8 (lanes 16-31),
// N = lane & 15.
// ---------------------------------------------------------------------------
__global__ __launch_bounds__(256) void edges_gemm_wmma_kernel(
    const float* __restrict__ A,   // global_attr [E_EDGES, DG]
    const float* __restrict__ B,   // edge_attr   [DG, FE]
    float* __restrict__ out)       // [E_EDGES, ROW_STRIDE]
{
    const unsigned waves_per_blk = blockDim.x >> 5;
    const unsigned wave = blockIdx.x * waves_per_blk + (threadIdx.x >> 5);
    const unsigned lane = threadIdx.x & 31;
    const unsigned waves_total = gridDim.x * waves_per_blk;
    const unsigned tiles = E_EDGES / 16;   // 16352

    const unsigned r = lane & 15;          // M (for A) / N (for B,C)
    const unsigned h = lane >> 4;          // K-pair selector

    // Preload all of B (2KB) into registers: breg[ntile][kstep] holds
    // { B[k0+2h][n], B[k0+2h+1][n] } with n = ntile*16 + r.
    v2f breg[2][4];
#pragma unroll
    for (int nt = 0; nt < 2; ++nt) {
#pragma unroll
        for (int ks = 0; ks < 4; ++ks) {
            const unsigned kb = (unsigned)ks * 4 + 2 * h;
            const unsigned n  = (unsigned)nt * 16 + r;
            breg[nt][ks].x = B[kb * FE + n];
            breg[nt][ks].y = B[(kb + 1) * FE + n];
        }
    }

    for (unsigned t = wave; t < tiles; t += waves_total) {
        const unsigned m0  = t * 16;
        const unsigned row = m0 + r;

        // A fragments: 4 float2 loads per lane (one per k-step)
        v2f areg[4];
#pragma unroll
        for (int ks = 0; ks < 4; ++ks) {
            areg[ks] = *(const v2f*)(A + (size_t)row * DG + (unsigned)ks * 4 + 2 * h);
        }

        v8f c0 = {};
        v8f c1 = {};
#pragma unroll
        for (int ks = 0; ks < 4; ++ks) {
            c0 = __builtin_amdgcn_wmma_f32_16x16x4_f32(
                false, areg[ks], false, breg[0][ks], (short)0, c0, false, false);
            c1 = __builtin_amdgcn_wmma_f32_16x16x4_f32(
                false, areg[ks], false, breg[1][ks], (short)0, c1, false, false);
        }

        // Scatter accumulators into out[:, 128:160]
#pragma unroll
        for (int v = 0; v < 8; ++v) {
            const unsigned m = m0 + (unsigned)v + 8 * h;
            float* dst = out + (size_t)m * ROW_STRIDE + 2 * F_DIM;
            __builtin_nontemporal_store(c0[v], dst + r);
            __builtin_nontemporal_store(c1[v], dst + 16 + r);
        }
    }
}

extern "C" void kernel_launch(void* const* d_in, const int* in_sizes, int n_in,
                              void* d_out, int out_size, void* d_ws, size_t ws_size,
                              hipStream_t stream) {
    (void)in_sizes; (void)n_in; (void)out_size; (void)d_ws; (void)ws_size;

    const float* nodes       = (const float*)d_in[0]; // [512, 64]
    const float* edge_attr   = (const float*)d_in[1]; // [16, 32]
    const float* global_attr = (const float*)d_in[2]; // [E, 16]
    float* out = (float*)d_out;                       // [E, 160]

    // Gather: one wave per edge, 8 waves (256 threads) per block.
    const int gather_blocks = E_EDGES / 8;            // 32704, E divisible by 8
    gather_kernel<<<gather_blocks, 256, 0, stream>>>(nodes, out);

    // GEMM: one wave per 16-edge tile, 8 tiles per block.
    const int tiles = E_EDGES / 16;                   // 16352
    const int gemm_blocks = tiles / 8;                // 2044, exact
    edges_gemm_wmma_kernel<<<gemm_blocks, 256, 0, stream>>>(global_attr, edge_attr, out);
}